// SGEncoder_91242285236450
// MI455X (gfx1250) — compile-verified
//
#include <hip/hip_runtime.h>
#include <math.h>

// ---------------------------------------------------------------------------
// Problem constants (fixed by setup_inputs): BS=64, N=10000, D=128, M=16, SG=1000
// ---------------------------------------------------------------------------
constexpr int kBS = 64;
constexpr int kN  = 10000;
constexpr int kD  = 128;
constexpr int kM  = 16;
constexpr int kSG = 1000;
constexpr int kSlices   = 8;               // node-slices per batch for the gather
constexpr int kSliceLen = kSG / kSlices;   // 125

typedef __attribute__((ext_vector_type(2))) float v2f;
typedef __attribute__((ext_vector_type(4))) float v4f;
typedef __attribute__((ext_vector_type(8))) float v8f;

__device__ __forceinline__ v8f wmma_f32_k4(v2f a, v2f b, v8f c) {
  // D = A(16x4 f32) * B(4x16 f32) + C(16x16 f32), exact f32 on the matrix pipe.
  return __builtin_amdgcn_wmma_f32_16x16x4_f32(
      /*neg_a=*/false, a, /*neg_b=*/false, b,
      /*c_mod=*/(short)0, c, /*reuse_a=*/false, /*reuse_b=*/false);
}

__device__ __forceinline__ float gelu_tanh(float x) {
  // jax.nn.gelu default (approximate=True)
  const float c = 0.7978845608028654f;   // sqrt(2/pi)
  float x3 = x * x * x;
  return 0.5f * x * (1.0f + tanhf(c * (x + 0.044715f * x3)));
}

// ---------------------------------------------------------------------------
// Kernel 1: gather + sum/max pooling partials.
// grid = kBS*kSlices blocks, 256 threads = 8 row-reader groups of 32 lanes.
// Each group reads entire 512B node rows with global_load_b128
// (lane covers dims 4*lane .. 4*lane+3).
// part_sum/part_max layout: [BS][kSlices][D]
// ---------------------------------------------------------------------------
__global__ __launch_bounds__(256)
void gather_pool_kernel(const float* __restrict__ node_emb,
                        const int*   __restrict__ sg_node_idx,
                        float* __restrict__ part_sum,
                        float* __restrict__ part_max) {
  __shared__ int   sidx[kSliceLen];
  __shared__ float rsum[8 * kD];
  __shared__ float rmax[8 * kD];

  const int tid   = threadIdx.x;
  const int lane  = tid & 31;
  const int grp   = tid >> 5;             // 0..7 row-reader group
  const int b     = blockIdx.x >> 3;
  const int slice = blockIdx.x & 7;

  if (tid < kSliceLen)
    sidx[tid] = sg_node_idx[b * kSG + slice * kSliceLen + tid];
  __syncthreads();

  const long base = (long)b * kN;

  float s0 = 0.f, s1 = 0.f, s2 = 0.f, s3 = 0.f;
  float m0 = -INFINITY, m1 = -INFINITY, m2 = -INFINITY, m3 = -INFINITY;

  for (int i = grp; i < kSliceLen; i += 8) {
    int idx = sidx[i];
    v4f v = {0.f, 0.f, 0.f, 0.f};
    if (idx != 0) {
      // 16B-aligned: row offset is a multiple of 512B, + lane*16
      v = *(const v4f*)(node_emb + (base + idx) * kD + lane * 4);
    }
    // node_emb[:,0] forced to zero in the reference -> idx==0 contributes 0.0
    s0 += v.x; s1 += v.y; s2 += v.z; s3 += v.w;
    m0 = fmaxf(m0, v.x); m1 = fmaxf(m1, v.y);
    m2 = fmaxf(m2, v.z); m3 = fmaxf(m3, v.w);
  }

  int o = grp * kD + lane * 4;
  rsum[o + 0] = s0; rsum[o + 1] = s1; rsum[o + 2] = s2; rsum[o + 3] = s3;
  rmax[o + 0] = m0; rmax[o + 1] = m1; rmax[o + 2] = m2; rmax[o + 3] = m3;
  __syncthreads();

  if (tid < kD) {
    float S  = 0.f;
    float Mx = -INFINITY;
    for (int g = 0; g < 8; ++g) {
      S += rsum[g * kD + tid];
      Mx = fmaxf(Mx, rmax[g * kD + tid]);
    }
    int oo = (b * kSlices + slice) * kD + tid;
    part_sum[oo] = S;
    part_max[oo] = Mx;
  }
}

// ---------------------------------------------------------------------------
// Kernel 2: sg_emb = mlp2_ln(pooled, sg_*) + mlp2_ln(meta, mf_*)
// grid = 4 blocks (M-tiles of 16 rows), 256 threads = 8 waves (N-tiles of 16 cols).
// ---------------------------------------------------------------------------
__global__ __launch_bounds__(256)
void sg_emb_kernel(const float* __restrict__ part_sum,
                   const float* __restrict__ part_max,
                   const float* __restrict__ meta,      // (64,16)
                   const float* __restrict__ sw1,       // (256,128)
                   const float* __restrict__ sb1,
                   const float* __restrict__ sg,
                   const float* __restrict__ sbt,
                   const float* __restrict__ sw2,       // (128,128)
                   const float* __restrict__ sb2,
                   const float* __restrict__ mw1,       // (16,128)
                   const float* __restrict__ mb1,
                   const float* __restrict__ mg,
                   const float* __restrict__ mbt,
                   const float* __restrict__ mw2,       // (128,128)
                   const float* __restrict__ mb2,
                   float* __restrict__ out)             // sg_emb (64,128)
{
  __shared__ float Abuf[16 * 256];   // pooled tile
  __shared__ float hA[16 * 128];     // sg hidden
  __shared__ float hB[16 * 128];     // mf hidden
  __shared__ float mu1[16], rs1[16], mu2[16], rs2[16];

  const int tid  = threadIdx.x;
  const int lane = tid & 31;
  const int wave = tid >> 5;
  const int row0 = blockIdx.x * 16;

  const int ml   = lane & 15;        // A row within tile / C col
  const int kh   = lane >> 4;        // K-half selector
  const int ncol = wave * 16 + ml;   // global output column

  // Build pooled tile: cols 0..127 = sum-reduce partials, 128..255 = max-reduce.
  for (int e = tid; e < 16 * 256; e += 256) {
    int r = e >> 8, k = e & 255;
    int b = row0 + r;
    float v;
    if (k < kD) {
      v = 0.0f;
      for (int s = 0; s < kSlices; ++s) v += part_sum[(b * kSlices + s) * kD + k];
    } else {
      int dd = k - kD;
      v = -INFINITY;
      for (int s = 0; s < kSlices; ++s) v = fmaxf(v, part_max[(b * kSlices + s) * kD + dd]);
    }
    Abuf[e] = v;
  }
  __syncthreads();

  // GEMM1a: pooled(16x256) @ sg_w1(:,ncol)   K=256 -> 64 wmma steps
  v8f acc1 = {};
  for (int kk = 0; kk < 64; ++kk) {
    int k0 = kk * 4;
    v2f a, bf;
    a.x  = Abuf[ml * 256 + k0 + 2 * kh];
    a.y  = Abuf[ml * 256 + k0 + 2 * kh + 1];
    bf.x = sw1[(k0 + 2 * kh) * kD + ncol];
    bf.y = sw1[(k0 + 2 * kh + 1) * kD + ncol];
    acc1 = wmma_f32_k4(a, bf, acc1);
  }
  // GEMM1b: meta(16x16) @ mf_w1(:,ncol)      K=16 -> 4 wmma steps
  v8f acc2 = {};
  for (int kk = 0; kk < 4; ++kk) {
    int k0 = kk * 4;
    v2f a, bf;
    a.x  = meta[(row0 + ml) * kM + k0 + 2 * kh];
    a.y  = meta[(row0 + ml) * kM + k0 + 2 * kh + 1];
    bf.x = mw1[(k0 + 2 * kh) * kD + ncol];
    bf.y = mw1[(k0 + 2 * kh + 1) * kD + ncol];
    acc2 = wmma_f32_k4(a, bf, acc2);
  }

  // Bias, spill tiles to LDS for row-wise LayerNorm stats.
  float bias1 = sb1[ncol], bias2 = mb1[ncol];
  for (int v = 0; v < 8; ++v) {
    int r = v + 8 * kh;
    hA[r * kD + ncol] = acc1[v] + bias1;
    hB[r * kD + ncol] = acc2[v] + bias2;
  }
  __syncthreads();

  if (tid < 16) {
    float s = 0.f, ss = 0.f;
    for (int cc = 0; cc < kD; ++cc) { float x = hA[tid * kD + cc]; s += x; ss += x * x; }
    float m = s * (1.0f / kD);
    mu1[tid] = m;
    rs1[tid] = rsqrtf(ss * (1.0f / kD) - m * m + 1e-5f);
  } else if (tid < 32) {
    int r = tid - 16;
    float s = 0.f, ss = 0.f;
    for (int cc = 0; cc < kD; ++cc) { float x = hB[r * kD + cc]; s += x; ss += x * x; }
    float m = s * (1.0f / kD);
    mu2[r] = m;
    rs2[r] = rsqrtf(ss * (1.0f / kD) - m * m + 1e-5f);
  }
  __syncthreads();

  // LN * gamma + beta, then GELU, in place.
  for (int e = tid; e < 16 * kD; e += 256) {
    int r = e >> 7, cc = e & (kD - 1);
    hA[e] = gelu_tanh((hA[e] - mu1[r]) * rs1[r] * sg[cc] + sbt[cc]);
    hB[e] = gelu_tanh((hB[e] - mu2[r]) * rs2[r] * mg[cc] + mbt[cc]);
  }
  __syncthreads();

  // GEMM2: hA @ sg_w2 + hB @ mf_w2  (K=128 each -> 32+32 wmma steps)
  v8f acc = {};
  for (int kk = 0; kk < 32; ++kk) {
    int k0 = kk * 4;
    v2f a, bf;
    a.x  = hA[ml * kD + k0 + 2 * kh];
    a.y  = hA[ml * kD + k0 + 2 * kh + 1];
    bf.x = sw2[(k0 + 2 * kh) * kD + ncol];
    bf.y = sw2[(k0 + 2 * kh + 1) * kD + ncol];
    acc = wmma_f32_k4(a, bf, acc);
  }
  for (int kk = 0; kk < 32; ++kk) {
    int k0 = kk * 4;
    v2f a, bf;
    a.x  = hB[ml * kD + k0 + 2 * kh];
    a.y  = hB[ml * kD + k0 + 2 * kh + 1];
    bf.x = mw2[(k0 + 2 * kh) * kD + ncol];
    bf.y = mw2[(k0 + 2 * kh + 1) * kD + ncol];
    acc = wmma_f32_k4(a, bf, acc);
  }
  float biaso = sb2[ncol] + mb2[ncol];
  for (int v = 0; v < 8; ++v) {
    int r = v + 8 * kh;
    out[(row0 + r) * kD + ncol] = acc[v] + biaso;
  }
}

// ---------------------------------------------------------------------------
// Kernel 3: ctxt = gelu(cat(sg_emb, graph_emb) @ cx_w1 + cx_b1) @ cx_w2 + cx_b2
// ---------------------------------------------------------------------------
__global__ __launch_bounds__(256)
void ctxt_kernel(const float* __restrict__ sg_emb,     // (64,128)
                 const float* __restrict__ graph_emb,  // (64,128)
                 const float* __restrict__ cw1,        // (256,128)
                 const float* __restrict__ cb1,
                 const float* __restrict__ cw2,        // (128,128)
                 const float* __restrict__ cb2,
                 float* __restrict__ out)              // ctxt_emb (64,128)
{
  __shared__ float Abuf[16 * 256];
  __shared__ float hA[16 * 128];

  const int tid  = threadIdx.x;
  const int lane = tid & 31;
  const int wave = tid >> 5;
  const int row0 = blockIdx.x * 16;

  const int ml   = lane & 15;
  const int kh   = lane >> 4;
  const int ncol = wave * 16 + ml;

  for (int e = tid; e < 16 * 256; e += 256) {
    int r = e >> 8, k = e & 255;
    Abuf[e] = (k < kD) ? sg_emb[(row0 + r) * kD + k]
                       : graph_emb[(row0 + r) * kD + (k - kD)];
  }
  __syncthreads();

  v8f acc1 = {};
  for (int kk = 0; kk < 64; ++kk) {
    int k0 = kk * 4;
    v2f a, bf;
    a.x  = Abuf[ml * 256 + k0 + 2 * kh];
    a.y  = Abuf[ml * 256 + k0 + 2 * kh + 1];
    bf.x = cw1[(k0 + 2 * kh) * kD + ncol];
    bf.y = cw1[(k0 + 2 * kh + 1) * kD + ncol];
    acc1 = wmma_f32_k4(a, bf, acc1);
  }
  float b1 = cb1[ncol];
  for (int v = 0; v < 8; ++v) {
    int r = v + 8 * kh;
    hA[r * kD + ncol] = gelu_tanh(acc1[v] + b1);
  }
  __syncthreads();

  v8f acc = {};
  for (int kk = 0; kk < 32; ++kk) {
    int k0 = kk * 4;
    v2f a, bf;
    a.x  = hA[ml * kD + k0 + 2 * kh];
    a.y  = hA[ml * kD + k0 + 2 * kh + 1];
    bf.x = cw2[(k0 + 2 * kh) * kD + ncol];
    bf.y = cw2[(k0 + 2 * kh + 1) * kD + ncol];
    acc = wmma_f32_k4(a, bf, acc);
  }
  float b2 = cb2[ncol];
  for (int v = 0; v < 8; ++v) {
    int r = v + 8 * kh;
    out[(row0 + r) * kD + ncol] = acc[v] + b2;
  }
}

// ---------------------------------------------------------------------------
extern "C" void kernel_launch(void* const* d_in, const int* in_sizes, int n_in,
                              void* d_out, int out_size, void* d_ws, size_t ws_size,
                              hipStream_t stream) {
  (void)in_sizes; (void)n_in; (void)out_size; (void)ws_size;

  const float* node_emb  = (const float*)d_in[0];
  const float* graph_emb = (const float*)d_in[1];
  const int*   sg_idx    = (const int*)  d_in[2];
  const float* meta      = (const float*)d_in[3];
  const float* sw1 = (const float*)d_in[4];
  const float* sb1 = (const float*)d_in[5];
  const float* sgm = (const float*)d_in[6];
  const float* sbt = (const float*)d_in[7];
  const float* sw2 = (const float*)d_in[8];
  const float* sb2 = (const float*)d_in[9];
  const float* mw1 = (const float*)d_in[10];
  const float* mb1 = (const float*)d_in[11];
  const float* mg  = (const float*)d_in[12];
  const float* mbt = (const float*)d_in[13];
  const float* mw2 = (const float*)d_in[14];
  const float* mb2 = (const float*)d_in[15];
  const float* cw1 = (const float*)d_in[16];
  const float* cb1 = (const float*)d_in[17];
  const float* cw2 = (const float*)d_in[18];
  const float* cb2 = (const float*)d_in[19];

  float* out_sg   = (float*)d_out;                 // (64,128)
  float* out_ctxt = (float*)d_out + kBS * kD;      // (64,128)

  float* part_sum = (float*)d_ws;                          // [64][8][128]
  float* part_max = part_sum + kBS * kSlices * kD;         // [64][8][128]

  gather_pool_kernel<<<kBS * kSlices, 256, 0, stream>>>(node_emb, sg_idx,
                                                        part_sum, part_max);
  sg_emb_kernel<<<kBS / 16, 256, 0, stream>>>(part_sum, part_max, meta,
                                              sw1, sb1, sgm, sbt, sw2, sb2,
                                              mw1, mb1, mg, mbt, mw2, mb2,
                                              out_sg);
  ctxt_kernel<<<kBS / 16, 256, 0, stream>>>(out_sg, graph_emb,
                                            cw1, cb1, cw2, cb2, out_ctxt);
}